// Routing_58274116272627
// MI455X (gfx1250) — compile-verified
//
#include <hip/hip_runtime.h>
#include <math.h>
#include <stdint.h>

#define B_TOT       64
#define IN_CAPS     2048
#define IN_DIM      16
#define NUM_CAPS    32
#define DIM_CAPS    16
#define NUM_ROUTING 3
#define EPS_SQ      1e-7f

#define NWAVES 8
#define BLOCK  (NWAVES * 32)
#define CI     32                 // i-chunk staged per TDM transfer
#define NCHUNK (IN_CAPS / CI)     // 64
#define IPW    (CI / NWAVES)      // 4 i per wave per chunk

typedef __attribute__((ext_vector_type(2))) float v2f;
typedef __attribute__((ext_vector_type(8))) float v8f;
typedef unsigned int tdm_v4u __attribute__((ext_vector_type(4)));
typedef int          tdm_v8i __attribute__((ext_vector_type(8)));
typedef int          tdm_v4i __attribute__((ext_vector_type(4)));

#if defined(__gfx1250__) && __has_builtin(__builtin_amdgcn_tensor_load_to_lds)
#define USE_TDM 1
#else
#define USE_TDM 0
#endif

// ---- LDS byte-offset map (single dynamic allocation, 260KB < 320KB WGP LDS) ----
#define OFF_LOGITS  0
#define LOGITS_B    (IN_CAPS * 16 * 4)        // 131072: logits[i][b]
#define OFF_SRED    (LOGITS_B)                // [16 b][16 m]
#define OFF_OUT     (OFF_SRED + 1024)         // [16 b][16 m]
#define OFF_RED     (OFF_OUT + 1024)          // [16 g][16 b]
#define OFF_INVD    (OFF_RED + 1024)          // [16 b]
#define OFF_SCALE   (OFF_INVD + 64)           // [16 b]
#define OFF_X       135168                    // 2 x 32KB X stage: [16 b][CI*16 f32]
#define XBUF_B      (16 * CI * IN_DIM * 4)    // 32768
#define OFF_W       (OFF_X + 2 * XBUF_B)      // 2 x 32KB W stage: [CI i][256 f32]
#define WBUF_B      (CI * IN_DIM * DIM_CAPS * 4)
#define SMEM_BYTES  (OFF_W + 2 * WBUF_B)      // 266240 B

#if USE_TDM
// Pack a 2D Tensor-DMA descriptor (D#) and issue TENSOR_LOAD_TO_LDS.
// Group0: count=1 | lds_addr | global_addr[56:0] | type=2("image").
// Group1: data_size=4B, tensor_dim = tile_dim (in-bounds), dim0 stride.
// This toolchain's builtin takes 6 args: (g0 v4u, g1 v8i, g2 v4i, g3 v4i,
// extra v8i, i32 cpol) — groups 2/3 and the extra group are zero (2D tensor).
__device__ __forceinline__ void tdm_load_2d(unsigned long long gaddr,
                                            unsigned lds_off,
                                            unsigned tile_d0,   // elems, contiguous
                                            unsigned tile_d1,   // rows
                                            unsigned long long stride0) // elems
{
    tdm_v4u g0;
    g0[0] = 1u;                                       // count=1, user descriptor
    g0[1] = lds_off;                                  // LDS byte address
    g0[2] = (unsigned)(gaddr & 0xffffffffull);
    g0[3] = (unsigned)((gaddr >> 32) & 0x01ffffffull) | (2u << 30);
    tdm_v8i g1;
    g1[0] = (int)(2u << 16);                          // data_size = 2 -> 4 bytes
    g1[1] = (int)((tile_d0 & 0xffffu) << 16);                                  // tensor_dim0 lo
    g1[2] = (int)(((tile_d0 >> 16) & 0xffffu) | ((tile_d1 & 0xffffu) << 16));  // d0 hi | d1 lo
    g1[3] = (int)(((tile_d1 >> 16) & 0xffffu) | ((tile_d0 & 0xffffu) << 16));  // d1 hi | tile_dim0
    g1[4] = (int)(tile_d1 & 0xffffu);                 // tile_dim1 | tile_dim2=0
    g1[5] = (int)(stride0 & 0xffffffffull);           // tensor_dim0_stride lo
    g1[6] = (int)((stride0 >> 32) & 0xffffull);       // stride hi | stride1=0
    g1[7] = 0;
    tdm_v4i gz4 = {0, 0, 0, 0};
    tdm_v8i gz8 = {0, 0, 0, 0, 0, 0, 0, 0};
    __builtin_amdgcn_tensor_load_to_lds(g0, g1, gz4, gz4, gz8, 0);
}

// Stage one i-chunk: X tile (16 batch rows x CI*16 contiguous f32, row stride
// IN_CAPS*16) and W tile (CI rows x 256 contiguous f32, row stride 32*256).
__device__ __forceinline__ void issue_chunk(const float* X, const float* W,
                                            int bt, int j, int c, int buf)
{
    const int ib = c * CI;
    unsigned long long xga = (unsigned long long)(uintptr_t)
        (X + ((size_t)(bt * 16) * IN_CAPS + ib) * IN_DIM);
    tdm_load_2d(xga, (unsigned)(OFF_X + buf * XBUF_B),
                CI * IN_DIM, 16, (unsigned long long)IN_CAPS * IN_DIM);
    unsigned long long wga = (unsigned long long)(uintptr_t)
        (W + ((size_t)ib * NUM_CAPS + j) * (IN_DIM * DIM_CAPS));
    tdm_load_2d(wga, (unsigned)(OFF_W + buf * WBUF_B),
                IN_DIM * DIM_CAPS, CI,
                (unsigned long long)NUM_CAPS * IN_DIM * DIM_CAPS);
}
#endif

// Fused capsule routing: grid (NUM_CAPS, B_TOT/16); one WG owns (j, 16-batch
// tile). inputs_hat is never materialized; W/x stream from the 192MB L2 via
// double-buffered TDM chunks into LDS and feed v_wmma_f32_16x16x4_f32 chains.
__global__ __launch_bounds__(BLOCK)
void caps_routing_kernel(const float* __restrict__ X,   // [64,2048,16]
                         const float* __restrict__ W,   // [2048,32,16,16]
                         float* __restrict__ Out)       // [64,32,16]
{
    extern __shared__ char smem_raw[];
    float* logits  = (float*)(smem_raw + OFF_LOGITS);
    float* s_red   = (float*)(smem_raw + OFF_SRED);
    float* out_t   = (float*)(smem_raw + OFF_OUT);
    float* red     = (float*)(smem_raw + OFF_RED);
    float* invd    = (float*)(smem_raw + OFF_INVD);
    float* scale_l = (float*)(smem_raw + OFF_SCALE);

    const int tid  = threadIdx.x;
    const int lane = tid & 31;
    const int wave = tid >> 5;
    const int half = lane >> 4;        // 0: lanes 0-15, 1: lanes 16-31
    const int col  = lane & 15;        // A row (=batch b) AND B/D column (=m)
    const int j    = blockIdx.x;
    const int bt   = blockIdx.y;

    for (int k = tid; k < IN_CAPS * 16; k += BLOCK) logits[k] = 0.f;
    __syncthreads();

    for (int r = 0; r < NUM_ROUTING; ++r) {
        // ---- softmax denominator over i for each b (axis=1 softmax) ----
        {
            const int b = tid & 15, g = tid >> 4;
            float p = 0.f;
            for (int i = g; i < IN_CAPS; i += 16)
                p += __expf(logits[i * 16 + b]);
            red[g * 16 + b] = p;
        }
        __syncthreads();
        if (tid < 16) {
            float d = 0.f;
            for (int g = 0; g < 16; ++g) d += red[g * 16 + tid];
            invd[tid] = 1.f / d;
        }
        __syncthreads();

        // ---- pass B: s = sum_i c_i * (X_i @ W_i) as one WMMA chain ----
        v8f acc = {};
        const float inv_d = invd[col];
#if USE_TDM
        if (wave == 0) issue_chunk(X, W, bt, j, 0, 0);
#endif
        for (int c = 0; c < NCHUNK; ++c) {
            const int buf = c & 1;
#if USE_TDM
            if (wave == 0) {
                if (c + 1 < NCHUNK) {
                    issue_chunk(X, W, bt, j, c + 1, buf ^ 1);   // prefetch next
                    __builtin_amdgcn_s_wait_tensorcnt((short)2); // chunk c landed
                } else {
                    __builtin_amdgcn_s_wait_tensorcnt((short)0);
                }
            }
            __syncthreads();   // publish TDM-written LDS to all waves
#else
            __syncthreads();
            {
                const int ib = c * CI;
                float* xb = (float*)(smem_raw + OFF_X + buf * XBUF_B);
                for (int k = tid; k < 16 * CI * IN_DIM; k += BLOCK) {
                    int b = k / (CI * IN_DIM), rem = k % (CI * IN_DIM);
                    xb[k] = X[((size_t)(bt * 16 + b) * IN_CAPS + ib) * IN_DIM + rem];
                }
                float* wb = (float*)(smem_raw + OFF_W + buf * WBUF_B);
                for (int k = tid; k < CI * 256; k += BLOCK) {
                    int il = k >> 8, rem = k & 255;
                    wb[k] = W[((size_t)(c * CI + il) * NUM_CAPS + j) * 256 + rem];
                }
                (void)ib;
            }
            __syncthreads();
#endif
            const float* xbuf = (const float*)(smem_raw + OFF_X + buf * XBUF_B);
            const float* wbuf = (const float*)(smem_raw + OFF_W + buf * WBUF_B);
            #pragma unroll
            for (int t = 0; t < IPW; ++t) {
                const int il = wave * IPW + t;
                const int i  = c * CI + il;
                const float cs = __expf(logits[i * 16 + col]) * inv_d;
                const float* xr = xbuf + col * (CI * IN_DIM) + il * IN_DIM;
                const float* wr = wbuf + il * 256;
                #pragma unroll
                for (int kc = 0; kc < 4; ++kc) {
                    const int n0 = kc * 4 + half * 2;   // A: K = v + 2*half
                    v2f a; a.x = xr[n0] * cs;       a.y = xr[n0 + 1] * cs;
                    v2f b; b.x = wr[n0 * 16 + col]; b.y = wr[(n0 + 1) * 16 + col];
                    acc = __builtin_amdgcn_wmma_f32_16x16x4_f32(
                            false, a, false, b, (short)0, acc, false, false);
                }
            }
            __syncthreads();   // buf free for the prefetch two chunks ahead
        }

        // reduce partial s across the 8 waves via LDS float atomics
        for (int k = tid; k < 256; k += BLOCK) s_red[k] = 0.f;
        __syncthreads();
        #pragma unroll
        for (int v = 0; v < 8; ++v) {
            const int b = v + 8 * half;                 // D row = batch
            atomicAdd(&s_red[b * 16 + col], acc[v]);    // D col = m
        }
        __syncthreads();

        // ---- squash along m ----
        if (tid < 16) {
            float sq = 0.f;
            for (int m = 0; m < 16; ++m) { float t = s_red[tid * 16 + m]; sq += t * t; }
            scale_l[tid] = sq / (1.f + sq) * rsqrtf(sq + EPS_SQ);
        }
        __syncthreads();
        for (int k = tid; k < 256; k += BLOCK)
            out_t[k] = scale_l[k >> 4] * s_red[k];
        __syncthreads();

        if (r < NUM_ROUTING - 1) {
            // ---- pass C: logits[b,i] += <out[b,:], hat[b,i,:]> ----
#if USE_TDM
            if (wave == 0) issue_chunk(X, W, bt, j, 0, 0);
#endif
            for (int c = 0; c < NCHUNK; ++c) {
                const int buf = c & 1;
#if USE_TDM
                if (wave == 0) {
                    if (c + 1 < NCHUNK) {
                        issue_chunk(X, W, bt, j, c + 1, buf ^ 1);
                        __builtin_amdgcn_s_wait_tensorcnt((short)2);
                    } else {
                        __builtin_amdgcn_s_wait_tensorcnt((short)0);
                    }
                }
                __syncthreads();
#else
                __syncthreads();
                {
                    float* xb = (float*)(smem_raw + OFF_X + buf * XBUF_B);
                    for (int k = tid; k < 16 * CI * IN_DIM; k += BLOCK) {
                        int b = k / (CI * IN_DIM), rem = k % (CI * IN_DIM);
                        xb[k] = X[((size_t)(bt * 16 + b) * IN_CAPS + c * CI) * IN_DIM + rem];
                    }
                    float* wb = (float*)(smem_raw + OFF_W + buf * WBUF_B);
                    for (int k = tid; k < CI * 256; k += BLOCK) {
                        int il = k >> 8, rem = k & 255;
                        wb[k] = W[((size_t)(c * CI + il) * NUM_CAPS + j) * 256 + rem];
                    }
                }
                __syncthreads();
#endif
                const float* xbuf = (const float*)(smem_raw + OFF_X + buf * XBUF_B);
                const float* wbuf = (const float*)(smem_raw + OFF_W + buf * WBUF_B);
                #pragma unroll
                for (int t = 0; t < IPW; ++t) {
                    const int il = wave * IPW + t;
                    const int i  = c * CI + il;
                    const float* xr = xbuf + col * (CI * IN_DIM) + il * IN_DIM;
                    const float* wr = wbuf + il * 256;
                    v8f hat = {};
                    #pragma unroll
                    for (int kc = 0; kc < 4; ++kc) {
                        const int n0 = kc * 4 + half * 2;
                        v2f a; a.x = xr[n0];            a.y = xr[n0 + 1];
                        v2f b; b.x = wr[n0 * 16 + col]; b.y = wr[(n0 + 1) * 16 + col];
                        hat = __builtin_amdgcn_wmma_f32_16x16x4_f32(
                                false, a, false, b, (short)0, hat, false, false);
                    }
                    #pragma unroll
                    for (int v = 0; v < 8; ++v) {
                        const int b = v + 8 * half;                 // D row = batch
                        float tt = hat[v] * out_t[b * 16 + col];    // * out[b, m=col]
                        tt += __shfl_xor(tt, 1, 16);                // sum over m
                        tt += __shfl_xor(tt, 2, 16);
                        tt += __shfl_xor(tt, 4, 16);
                        tt += __shfl_xor(tt, 8, 16);
                        if (col == 0) logits[i * 16 + b] += tt;     // unique writer
                    }
                }
                __syncthreads();
            }
        }
    }

    // ---- write outputs [b, j, m] ----
    for (int k = tid; k < 256; k += BLOCK) {
        const int b = k >> 4, m = k & 15;
        Out[(((size_t)(bt * 16 + b)) * NUM_CAPS + j) * DIM_CAPS + m] = out_t[k];
    }
}

extern "C" void kernel_launch(void* const* d_in, const int* in_sizes, int n_in,
                              void* d_out, int out_size, void* d_ws, size_t ws_size,
                              hipStream_t stream) {
    (void)in_sizes; (void)n_in; (void)d_ws; (void)ws_size; (void)out_size;
    const float* X = (const float*)d_in[0];   // inputs [64,2048,16] f32
    const float* W = (const float*)d_in[1];   // W [2048,32,16,16] f32
    float* Out = (float*)d_out;               // [64,32,16] f32

    dim3 grid(NUM_CAPS, B_TOT / 16);   // 32 x 4 = 128 workgroups
    dim3 block(BLOCK);                 // 8 wave32 waves
    caps_routing_kernel<<<grid, block, (size_t)SMEM_BYTES, stream>>>(X, W, Out);
}